// TokenRoutedMLP_17506286698736
// MI455X (gfx1250) — compile-verified
//
#include <hip/hip_runtime.h>
#include <hip/hip_bf16.h>

// ---------------------------------------------------------------------------
// Token-routed MoE MLP (SwiGLU) for MI455X (gfx1250), f32 end-to-end using
// V_WMMA_F32_16X16X4_F32 (exact f32 matrix MACs, matches f32 reference).
//
// Roofline: 206 GFLOP, 805MB f32 weights streamed => ~35us HBM floor at
// 23.3 TB/s. Grouped GEMM over expert-gathered token lists.
// Round-3 fix: B staged in LDS in WMMA-fragment-ready K-pair-interleaved
// layout (Bp[k>>1][col*2 + (k&1)]) so every B fragment is a single aligned
// ds_load_b64 feeding v_wmma directly -- removes the v_dual_mov shuffles.
// Pitch 288 dwords (== 32 mod 64 banks) keeps the two half-wave reads on
// disjoint bank halves.
// ---------------------------------------------------------------------------

#define T_TOK 4096
#define HID   2048
#define INTER 4096
#define NEXP  8
#define TWO_I 8192
#define VOCAB 50257
#define KC    32
#define BPITCH 288   // dwords per K-pair row (128 cols * 2 = 256 used)

typedef __attribute__((ext_vector_type(2))) float v2f;
typedef __attribute__((ext_vector_type(8))) float v8f;

// Workspace layout (4-byte units):
//   [0..7]                    counts[NEXP]
//   [16 .. 16+NEXP*T_TOK)     per-expert token lists
//   [WS_ACT_OFF ...)          act[T_TOK][INTER]  (~64MB f32)
#define WS_COUNTS_OFF 0
#define WS_LISTS_OFF  16
#define WS_ACT_OFF    (WS_LISTS_OFF + NEXP * T_TOK)

// ---------------------------------------------------------------------------
__global__ __launch_bounds__(256)
void moe_route(const int* __restrict__ token_ids,
               int* __restrict__ counts,
               int* __restrict__ lists) {
  if (threadIdx.x < NEXP) counts[threadIdx.x] = 0;
  __syncthreads();
  for (int t = (int)threadIdx.x; t < T_TOK; t += 256) {
    int tid = token_ids[t];
    tid = tid < 0 ? 0 : (tid > (VOCAB - 1) ? (VOCAB - 1) : tid);
    int e = tid & (NEXP - 1);
    int pos = atomicAdd(&counts[e], 1);
    lists[e * T_TOK + pos] = t;
  }
}

// ---------------------------------------------------------------------------
// Kernel 2: act[token, n] = silu(x@Wg[e])[n] * (x@Wu[e])[n]
// Block tile 32 tokens x 128 columns; 8 waves (2x4); wave tile 16x32 with
// 4 f32 accumulators (gate/up x 2 N-subtiles). K staged via LDS, 32/step.
// ---------------------------------------------------------------------------
__global__ __launch_bounds__(256)
void moe_gemm1_swiglu(const float* __restrict__ x,
                      const float* __restrict__ gup,   // [E][HID][2*INTER]
                      const int*  __restrict__ counts,
                      const int*  __restrict__ lists,
                      float* __restrict__ act) {       // [T_TOK][INTER]
  const int e     = blockIdx.x >> 7;     // 128 worst-case m-tiles per expert
  const int mt    = blockIdx.x & 127;
  const int count = counts[e];
  if (mt * 32 >= count) return;

  __shared__ __align__(16) float As[32][36];
  __shared__ __align__(16) float Bg[KC / 2][BPITCH];   // K-pair interleaved
  __shared__ __align__(16) float Bu[KC / 2][BPITCH];
  __shared__ int tokRow[32];

  const int t = (int)threadIdx.x;
  if (t < 32) {
    int gr = mt * 32 + t;
    tokRow[t] = (gr < count) ? lists[e * T_TOK + gr] : -1;
  }
  __syncthreads();

  // Hoisted, branch-free A gather: this thread always loads row `arow`.
  const int arow = t >> 3;           // 0..31
  const int acol = (t & 7) * 4;      // 0,4,...,28
  const int myTok = tokRow[arow];
  const float* aSrc = x + (size_t)(myTok < 0 ? 0 : myTok) * HID + acol;

  const int nbase = blockIdx.y * 128;
  const float* gupE = gup + (size_t)e * HID * TWO_I + nbase;

  const int wave = t >> 5;
  const int lane = t & 31;
  const int wm   = wave >> 2;        // 0..1
  const int wn   = wave & 3;         // 0..3
  const int lh   = lane >> 4;        // 0..1
  const int l15  = lane & 15;

  // B staging decomposition for this thread (fixed K row per i-slice)
  v8f cg[2] = {{}, {}};
  v8f cu[2] = {{}, {}};

  for (int kk = 0; kk < HID; kk += KC) {
    // stage A 32x32: one b128 per thread
    float4 av = *(const float4*)(aSrc + kk);
    *(float4*)&As[arow][acol] = av;
    // stage B gate/up 32x128 each, re-packed into K-pair interleaved layout
#pragma unroll
    for (int i = 0; i < 4; ++i) {
      int idx = t + i * 256;
      int br = idx >> 5;               // k row 0..31
      int bc = (idx & 31) * 4;         // col 0..124
      const float* src = gupE + (size_t)(kk + br) * TWO_I + bc;
      float4 g = *(const float4*)(src);
      float4 u = *(const float4*)(src + INTER);
      const int p  = br >> 1;          // K-pair row
      const int pa = (br & 1);         // parity within pair
#pragma unroll
      for (int j = 0; j < 4; ++j) {
        Bg[p][(bc + j) * 2 + pa] = ((const float*)&g)[j];
        Bu[p][(bc + j) * 2 + pa] = ((const float*)&u)[j];
      }
    }
    __syncthreads();

#pragma unroll
    for (int kg = 0; kg < 8; ++kg) {
      const int kb = kg * 4;
      const int p  = kg * 2 + lh;      // this lane's K-pair row
      v2f a;
      // A 16x4 layout: lane l holds A[l%16][2*(l/16)+v] (contiguous -> b64)
      a.x = As[wm * 16 + l15][kb + 2 * lh + 0];
      a.y = As[wm * 16 + l15][kb + 2 * lh + 1];
#pragma unroll
      for (int s = 0; s < 2; ++s) {
        const int col2 = (wn * 32 + s * 16 + l15) * 2;
        v2f bg = *(const v2f*)&Bg[p][col2];   // single aligned ds_load_b64
        v2f bu = *(const v2f*)&Bu[p][col2];
        cg[s] = __builtin_amdgcn_wmma_f32_16x16x4_f32(false, a, false, bg,
                                                      (short)0, cg[s], false, false);
        cu[s] = __builtin_amdgcn_wmma_f32_16x16x4_f32(false, a, false, bu,
                                                      (short)0, cu[s], false, false);
      }
    }
    __syncthreads();
  }

  // SwiGLU epilogue (C/D layout: VGPR r -> row r+8*(l/16), col l%16)
#pragma unroll
  for (int r = 0; r < 8; ++r) {
    const int m   = wm * 16 + r + 8 * lh;
    const int tok = tokRow[m];
    if (tok >= 0) {
#pragma unroll
      for (int s = 0; s < 2; ++s) {
        float g = cg[s][r], u = cu[s][r];
        float sig = g / (1.0f + __expf(-g));
        act[(size_t)tok * INTER + nbase + wn * 32 + s * 16 + l15] = sig * u;
      }
    }
  }
}

// ---------------------------------------------------------------------------
// Kernel 3: out[token, h] = act[token, :] @ down[e][:, h], K = INTER = 4096
// ---------------------------------------------------------------------------
__global__ __launch_bounds__(256)
void moe_gemm2_scatter(const float* __restrict__ act,   // [T_TOK][INTER]
                       const float* __restrict__ down,  // [E][INTER][HID]
                       const int*  __restrict__ counts,
                       const int*  __restrict__ lists,
                       float* __restrict__ out) {       // [T_TOK][HID]
  const int e     = blockIdx.x >> 7;
  const int mt    = blockIdx.x & 127;
  const int count = counts[e];
  if (mt * 32 >= count) return;

  __shared__ __align__(16) float As[32][36];
  __shared__ __align__(16) float Bs[KC / 2][BPITCH];
  __shared__ int tokRow[32];

  const int t = (int)threadIdx.x;
  if (t < 32) {
    int gr = mt * 32 + t;
    tokRow[t] = (gr < count) ? lists[e * T_TOK + gr] : -1;
  }
  __syncthreads();

  const int arow = t >> 3;
  const int acol = (t & 7) * 4;
  const int myTok = tokRow[arow];
  const float* aSrc = act + (size_t)(myTok < 0 ? 0 : myTok) * INTER + acol;

  const int hbase = blockIdx.y * 128;
  const float* dnE = down + (size_t)e * INTER * HID + hbase;

  const int wave = t >> 5;
  const int lane = t & 31;
  const int wm   = wave >> 2;
  const int wn   = wave & 3;
  const int lh   = lane >> 4;
  const int l15  = lane & 15;

  v8f acc[2] = {{}, {}};

  for (int kk = 0; kk < INTER; kk += KC) {
    float4 av = *(const float4*)(aSrc + kk);
    *(float4*)&As[arow][acol] = av;
#pragma unroll
    for (int i = 0; i < 4; ++i) {
      int idx = t + i * 256;
      int br = idx >> 5;
      int bc = (idx & 31) * 4;
      float4 b = *(const float4*)(dnE + (size_t)(kk + br) * HID + bc);
      const int p  = br >> 1;
      const int pa = (br & 1);
#pragma unroll
      for (int j = 0; j < 4; ++j) {
        Bs[p][(bc + j) * 2 + pa] = ((const float*)&b)[j];
      }
    }
    __syncthreads();

#pragma unroll
    for (int kg = 0; kg < 8; ++kg) {
      const int kb = kg * 4;
      const int p  = kg * 2 + lh;
      v2f a;
      a.x = As[wm * 16 + l15][kb + 2 * lh + 0];
      a.y = As[wm * 16 + l15][kb + 2 * lh + 1];
#pragma unroll
      for (int s = 0; s < 2; ++s) {
        const int col2 = (wn * 32 + s * 16 + l15) * 2;
        v2f b = *(const v2f*)&Bs[p][col2];
        acc[s] = __builtin_amdgcn_wmma_f32_16x16x4_f32(false, a, false, b,
                                                       (short)0, acc[s], false, false);
      }
    }
    __syncthreads();
  }

#pragma unroll
  for (int r = 0; r < 8; ++r) {
    const int m   = wm * 16 + r + 8 * lh;
    const int tok = tokRow[m];
    if (tok >= 0) {
#pragma unroll
      for (int s = 0; s < 2; ++s) {
        out[(size_t)tok * HID + hbase + wn * 32 + s * 16 + l15] = acc[s][r];
      }
    }
  }
}

// ---------------------------------------------------------------------------
extern "C" void kernel_launch(void* const* d_in, const int* in_sizes, int n_in,
                              void* d_out, int out_size, void* d_ws, size_t ws_size,
                              hipStream_t stream) {
  (void)in_sizes; (void)n_in; (void)out_size; (void)ws_size;

  const float* x         = (const float*)d_in[0];   // [T, H]
  const int*   token_ids = (const int*)d_in[1];     // [T]
  const float* gup       = (const float*)d_in[2];   // [E, H, 2I]
  const float* down      = (const float*)d_in[3];   // [E, I, H]
  float*       out       = (float*)d_out;           // [T, H]

  int*   wsi    = (int*)d_ws;
  int*   counts = wsi + WS_COUNTS_OFF;
  int*   lists  = wsi + WS_LISTS_OFF;
  float* act    = (float*)(wsi + WS_ACT_OFF);       // 64MB scratch

  moe_route<<<1, 256, 0, stream>>>(token_ids, counts, lists);

  // Worst case: all 4096 tokens on one expert -> 128 m-tiles of 32;
  // dead tiles exit immediately on counts[e].
  moe_gemm1_swiglu<<<dim3(NEXP * 128, INTER / 128), 256, 0, stream>>>(
      x, gup, counts, lists, act);

  moe_gemm2_scatter<<<dim3(NEXP * 128, HID / 128), 256, 0, stream>>>(
      act, down, counts, lists, out);
}